// ProductionStorage_37855841746970
// MI455X (gfx1250) — compile-verified
//
#include <hip/hip_runtime.h>
#include <hip/hip_bf16.h>
#include <stdint.h>

// ---------------------------------------------------------------------------
// GR4J production-store scan, split into:
//   K1 (parallel, BW-bound): leaky_relu + tanh precompute (native v_tanh_f32);
//       writes out cols 0-3, stashes (th_p, th_e) in cols 4-5 of the output.
//   K2 (serial, one wave32): state recurrence only (tanh hoisted out);
//       (th_p,th_e) pairs streamed global->LDS with GLOBAL_LOAD_ASYNC_TO_LDS_B64
//       (double buffered, s_wait_asynccnt fenced).
//   K3 (parallel, BW-bound): recompute p_s / perc from s_{t-1}, overwrite 4-5.
// Output layout (reference returns a tuple): [T,6] matrix flat, then [T] s_new.
// ---------------------------------------------------------------------------

#define SCALE_F   1000.0f
#define S_INIT_F  0.5f
#define NEG_SLOPE 0.01f
#define CHUNK     256
#define PER_LANE  (CHUNK / 32)

#if defined(__HIP_DEVICE_COMPILE__) && \
    __has_builtin(__builtin_amdgcn_global_load_async_to_lds_b64)
#define USE_ASYNC_LDS 1
#else
#define USE_ASYNC_LDS 0
#endif

#if USE_ASYNC_LDS
typedef int v2i __attribute__((vector_size(8)));
typedef __attribute__((address_space(1))) v2i* gv2i_ptr;   // global b64 payload
typedef __attribute__((address_space(3))) v2i* lv2i_ptr;   // LDS b64 payload

__device__ __forceinline__ void async_ld_b64(const float* gsrc, float2* ldst) {
    __builtin_amdgcn_global_load_async_to_lds_b64(
        (gv2i_ptr)(void*)gsrc, (lv2i_ptr)(void*)ldst, 0, 0);
}
#endif

// Native gfx1250 tanh (V_TANH_F32) when the toolchain exposes it.
__device__ __forceinline__ float fast_tanh(float x) {
#if defined(__HIP_DEVICE_COMPILE__) && __has_builtin(__builtin_amdgcn_tanhf)
    return __builtin_amdgcn_tanhf(x);
#else
    return tanhf(x);
#endif
}

struct StepOut { float ps, perc, snew; };

// One GR4J step given state s and precomputed tanh terms. Fast-path hardware
// rcp/rsq/sqrt keep the serial dependence chain short (no IEEE div expansion).
__device__ __forceinline__ StepOut gr4j_step(float s, float thp, float the,
                                             float x1s, float inv_x1s) {
    float r   = s * inv_x1s;
    float ps  = x1s * (1.0f - r * r) * thp *
                __builtin_amdgcn_rcpf(fmaf(r, thp, 1.0f));
    float es  = s * (2.0f - r) * the *
                __builtin_amdgcn_rcpf(fmaf(1.0f - r, the, 1.0f));
    float tmp = s + ps - es;
    float u   = (4.0f / 9.0f) * tmp * inv_x1s;
    float u2  = u * u;
    float u4  = u2 * u2;
    // (1+u^4)^(-1/4) == rsqrt(sqrt(1+u^4))
    float f   = __builtin_amdgcn_rsqf(__builtin_amdgcn_sqrtf(1.0f + u4));
    StepOut o;
    o.snew = tmp * f;
    o.perc = tmp - o.snew;
    o.ps   = ps;
    return o;
}

// ---------------------------------------------------------------------------
// K1: per-step input transform (fully parallel, ~90 MB traffic -> ~4us @23.3TB/s)
// ---------------------------------------------------------------------------
__global__ __launch_bounds__(256) void k_prep(const float* __restrict__ x,
                                              const float* __restrict__ x1,
                                              float* __restrict__ out6,
                                              int T) {
    int t = blockIdx.x * blockDim.x + threadIdx.x;
    if (t >= T) return;
    float x1s = x1[0] * SCALE_F;
    float inv = __builtin_amdgcn_rcpf(x1s);

    float2 in = ((const float2*)x)[t];          // [precip, pet], 8B aligned
    float pd  = in.x - in.y;
    float pn  = (pd >= 0.0f) ? pd : NEG_SLOPE * pd;    // leaky_relu(pd)
    float npd = -pd;
    float en  = (npd >= 0.0f) ? npd : NEG_SLOPE * npd; // leaky_relu(-pd)
    float thp = fast_tanh(pn * inv);
    float the = fast_tanh(en * inv);

    float2* row = (float2*)(out6 + (size_t)t * 6);     // 24B row, 8B aligned
    row[0] = in;                       // cols 0,1: precip, pet
    row[1] = make_float2(pn, en);      // cols 2,3: p_n, e_n
    row[2] = make_float2(thp, the);    // cols 4,5: tanh terms (temporary)
}

// ---------------------------------------------------------------------------
// K2: the irreducible serial recurrence. One wave32; (thp,the) pairs staged
// into LDS ahead of the consumer via async-to-LDS (ASYNCcnt tracked).
// ---------------------------------------------------------------------------
__global__ __launch_bounds__(32) void k_scan(const float* __restrict__ out6,
                                             float* __restrict__ sOut,
                                             const float* __restrict__ x1,
                                             int T) {
    __shared__ float2 sbuf[2][CHUNK];
    const int lane = threadIdx.x;                 // wave32: lane == threadIdx.x
    const float x1s = x1[0] * SCALE_F;
    const float inv = __builtin_amdgcn_rcpf(x1s);
    float s = S_INIT_F * x1s;
    const int nchunk = (T + CHUNK - 1) / CHUNK;

#if USE_ASYNC_LDS
    // --- async-to-LDS double-buffered pipeline ---
    {
        #pragma unroll
        for (int k = 0; k < PER_LANE; ++k) {
            int idx = k * 32 + lane;
            if (idx < T)
                async_ld_b64(out6 + (size_t)idx * 6 + 4, &sbuf[0][idx]);
        }
    }
    for (int c = 0; c < nchunk; ++c) {
        asm volatile("s_wait_asynccnt 0" ::: "memory");  // chunk c landed in LDS
        __syncthreads();   // DS reads of the buffer we are about to refill done
        if (c + 1 < nchunk) {
            int base = (c + 1) * CHUNK;
            float2* dst = sbuf[(c + 1) & 1];
            #pragma unroll
            for (int k = 0; k < PER_LANE; ++k) {
                int idx = k * 32 + lane;
                int t   = base + idx;
                if (t < T)
                    async_ld_b64(out6 + (size_t)t * 6 + 4, &dst[idx]);
            }
        }
        const float2* buf = sbuf[c & 1];
        const int base = c * CHUNK;
        const int nn = (T - base < CHUNK) ? (T - base) : CHUNK;
        for (int i = 0; i < nn; ++i) {
            float2 th2 = buf[i];                   // same addr all lanes: bcast
            StepOut o = gr4j_step(s, th2.x, th2.y, x1s, inv);
            s = o.snew;
            if (lane == 0) sOut[base + i] = s;     // fire-and-forget store
        }
    }
#else
    // --- fallback: register double-buffer (loads issued a chunk ahead) ---
    float2 regs[PER_LANE];
    {
        #pragma unroll
        for (int k = 0; k < PER_LANE; ++k) {
            int t = k * 32 + lane;
            regs[k] = (t < T) ? *(const float2*)(out6 + (size_t)t * 6 + 4)
                              : make_float2(0.0f, 0.0f);
        }
    }
    for (int c = 0; c < nchunk; ++c) {
        float2* buf = sbuf[c & 1];
        #pragma unroll
        for (int k = 0; k < PER_LANE; ++k) buf[k * 32 + lane] = regs[k];
        __syncthreads();
        if (c + 1 < nchunk) {
            int base = (c + 1) * CHUNK;
            #pragma unroll
            for (int k = 0; k < PER_LANE; ++k) {
                int t = base + k * 32 + lane;
                regs[k] = (t < T) ? *(const float2*)(out6 + (size_t)t * 6 + 4)
                                  : make_float2(0.0f, 0.0f);
            }
        }
        const int base = c * CHUNK;
        const int nn = (T - base < CHUNK) ? (T - base) : CHUNK;
        const float2* cbuf = sbuf[c & 1];
        for (int i = 0; i < nn; ++i) {
            float2 th2 = cbuf[i];
            StepOut o = gr4j_step(s, th2.x, th2.y, x1s, inv);
            s = o.snew;
            if (lane == 0) sOut[base + i] = s;
        }
        __syncthreads();
    }
#endif
}

// ---------------------------------------------------------------------------
// K3: recompute p_s/perc per step from s_{t-1} (fully parallel), overwrite 4-5.
// ---------------------------------------------------------------------------
__global__ __launch_bounds__(256) void k_outputs(float* __restrict__ out6,
                                                 const float* __restrict__ sOut,
                                                 const float* __restrict__ x1,
                                                 int T) {
    int t = blockIdx.x * blockDim.x + threadIdx.x;
    if (t >= T) return;
    float x1s = x1[0] * SCALE_F;
    float inv = __builtin_amdgcn_rcpf(x1s);
    float s   = (t == 0) ? S_INIT_F * x1s : sOut[t - 1];
    float2 th2 = *(const float2*)(out6 + (size_t)t * 6 + 4);
    StepOut o = gr4j_step(s, th2.x, th2.y, x1s, inv);
    *(float2*)(out6 + (size_t)t * 6 + 4) = make_float2(o.ps, o.perc);
}

// ---------------------------------------------------------------------------
extern "C" void kernel_launch(void* const* d_in, const int* in_sizes, int n_in,
                              void* d_out, int out_size, void* d_ws, size_t ws_size,
                              hipStream_t stream) {
    const float* x  = (const float*)d_in[0];   // [T,2]
    const float* x1 = (const float*)d_in[1];   // [1]
    const int T = in_sizes[0] / 2;

    float* out6 = (float*)d_out;                       // [T,6] flat
    float* sOut = (float*)d_out + (size_t)T * 6;       // [T]

    const int threads = 256;
    const int blocks  = (T + threads - 1) / threads;

    k_prep   <<<blocks, threads, 0, stream>>>(x, x1, out6, T);
    k_scan   <<<1, 32, 0, stream>>>(out6, sOut, x1, T);
    k_outputs<<<blocks, threads, 0, stream>>>(out6, sOut, x1, T);
}